// S5Layer_91242285236572
// MI455X (gfx1250) — compile-verified
//
#include <hip/hip_runtime.h>

// Problem constants (match reference)
static constexpr int HN   = 128;   // N_HIDDEN
static constexpr int KN   = 1024;  // N_IN (K of GEMM, also inner u dim)
static constexpr int TT   = 1024;  // sequence length (= u_input.shape[2])
static constexpr int BBAT = 2;     // batch
static constexpr int NCOL = TT * BBAT;  // GEMM N dimension (t*2 + b)
static constexpr int TC     = 128; // scan chunk length
static constexpr int NCHUNK = TT / TC;  // 8

typedef float v2f __attribute__((ext_vector_type(2)));
typedef float v8f __attribute__((ext_vector_type(8)));

// ---------------------------------------------------------------------------
// Kernel 1: ZOH discretization.  lambda_bar = exp(lam*delta),
// b_bar = ((lambda_bar-1)/lam) * (b_re + i b_im), split into re/im planes.
// ---------------------------------------------------------------------------
__global__ __launch_bounds__(256) void s5_precompute(
    const float* __restrict__ delta_p,
    const float* __restrict__ lre, const float* __restrict__ lim,
    const float* __restrict__ bre, const float* __restrict__ bim,
    float* __restrict__ bbr, float* __restrict__ bbi,
    float2* __restrict__ lambda_bar)
{
    int idx = blockIdx.x * blockDim.x + threadIdx.x;
    if (idx >= HN * KN) return;
    int h = idx / KN;
    int i = idx - h * KN;

    float d  = delta_p[0];
    float lr = lre[h];
    float li = lim[h];

    float e   = expf(lr * d);
    float lbr = e * cosf(li * d);
    float lbi = e * sinf(li * d);

    // c = (lambda_bar - 1) / lam   (complex division)
    float denom = lr * lr + li * li;
    float nr = lbr - 1.0f;
    float cr = (nr * lr + lbi * li) / denom;
    float ci = (lbi * lr - nr * li) / denom;

    float br = bre[idx], bi = bim[idx];
    bbr[idx] = cr * br - ci * bi;
    bbi[idx] = cr * bi + ci * br;

    if (i == 0) lambda_bar[h] = make_float2(lbr, lbi);
}

// ---------------------------------------------------------------------------
// Kernel 2: bu[n, h] (n = t*2 + b) via f32 WMMA.  One wave per 16x16 tile,
// two accumulators (re / im) sharing the real-valued B fragment from u.
//   bu_re = b_bar_re @ u^T ,  bu_im = b_bar_im @ u^T
// Fragment layouts per ISA 7.12.2 (32-bit, wave32):
//   A 16x4 : lane m = L&15 ; VGPR0/1 hold K = 2*(L>>4) + {0,1}
//   B 4x16 : lane n = L&15 ; VGPR0/1 hold K = 2*(L>>4) + {0,1}
//   D 16x16: VGPR j -> M = j + 8*(L>>4), N = L&15
// ---------------------------------------------------------------------------
__global__ __launch_bounds__(32) void s5_bu_gemm(
    const float* __restrict__ Are,   // b_bar_re [HN][KN]
    const float* __restrict__ Aim,   // b_bar_im [HN][KN]
    const float* __restrict__ U,     // u [BBAT][TT][KN]
    float2* __restrict__ bu)         // [NCOL][HN] complex
{
    int lane = threadIdx.x & 31;
    int mt   = blockIdx.y;           // 0..HN/16-1
    int nt   = blockIdx.x;           // 0..NCOL/16-1
    int half = lane >> 4;            // 0 or 1
    int ml   = lane & 15;

    int row = mt * 16 + ml;          // h for A loads
    int col = nt * 16 + ml;          // n for B loads: t = col>>1, b = col&1
    const float* arow = Are + (size_t)row * KN;
    const float* irow = Aim + (size_t)row * KN;
    const float* ucol = U + ((size_t)(col & 1) * TT + (size_t)(col >> 1)) * KN;

    v8f accr = {};
    v8f acci = {};
    int koff = 2 * half;
    for (int k = 0; k < KN; k += 4) {
        v2f a_re = *(const v2f*)(arow + k + koff);
        v2f a_im = *(const v2f*)(irow + k + koff);
        v2f bfr  = *(const v2f*)(ucol + k + koff);
        accr = __builtin_amdgcn_wmma_f32_16x16x4_f32(
            false, a_re, false, bfr, (short)0, accr, false, false);
        acci = __builtin_amdgcn_wmma_f32_16x16x4_f32(
            false, a_im, false, bfr, (short)0, acci, false, false);
    }

    int n = nt * 16 + ml;            // output column for this lane
    int hbase = mt * 16 + 8 * half;  // M = j + 8*half
#pragma unroll
    for (int j = 0; j < 8; ++j) {
        bu[(size_t)n * HN + hbase + j] = make_float2(accr[j], acci[j]);
    }
}

// ---------------------------------------------------------------------------
// Kernel 3: run the diagonal recurrence once (no stores) to capture the state
// at each chunk boundary:  bnd[c] = x_{c*TC - 1},  c = 1..NCHUNK-1.
// x_t[b,h1,h2] = LAM[h2]*x_{t-1}[b,h1,h2] + bu[t,b,h1]
// ---------------------------------------------------------------------------
__global__ __launch_bounds__(HN) void s5_scan_boundaries(
    const float2* __restrict__ lambda_bar,
    const float2* __restrict__ bu,      // [t*2+b][h1]
    float2* __restrict__ bnd)           // [NCHUNK][BBAT][HN][HN]
{
    int h2 = threadIdx.x;
    int bh = blockIdx.x;                // b*HN + h1
    int b  = bh >> 7;
    int h1 = bh & (HN - 1);
    float2 lam = lambda_bar[h2];
    float xr = 0.0f, xi = 0.0f;

    for (int t = 0; t < (NCHUNK - 1) * TC; ++t) {
        float2 buv = bu[(size_t)(t * BBAT + b) * HN + h1];  // broadcast load
        float nr = fmaf(lam.x, xr, fmaf(-lam.y, xi, buv.x));
        float ni = fmaf(lam.x, xi, fmaf( lam.y, xr, buv.y));
        xr = nr; xi = ni;
        if (((t + 1) & (TC - 1)) == 0) {
            int c = (t + 1) / TC;       // 1..NCHUNK-1
            bnd[(((size_t)c * BBAT + b) * HN + h1) * HN + h2] =
                make_float2(xr, xi);
        }
    }
}

// ---------------------------------------------------------------------------
// Kernel 4: per-chunk scan with output stores (the bandwidth-bound phase).
// 2048 blocks x 128 lanes = 8192 waves, each storing contiguous 256B per step.
// ---------------------------------------------------------------------------
__global__ __launch_bounds__(HN) void s5_scan_write(
    const float2* __restrict__ lambda_bar,
    const float2* __restrict__ bu,
    const float2* __restrict__ bnd,
    float2* __restrict__ out)           // [TT][BBAT][HN][HN]
{
    int h2 = threadIdx.x;
    int bh = blockIdx.x;                // b*HN + h1
    int c  = blockIdx.y;                // chunk
    int b  = bh >> 7;
    int h1 = bh & (HN - 1);
    float2 lam = lambda_bar[h2];

    float xr = 0.0f, xi = 0.0f;
    if (c > 0) {
        float2 v = bnd[(((size_t)c * BBAT + b) * HN + h1) * HN + h2];
        xr = v.x; xi = v.y;
    }

    int t0 = c * TC;
    for (int t = t0; t < t0 + TC; ++t) {
        float2 buv = bu[(size_t)(t * BBAT + b) * HN + h1];
        float nr = fmaf(lam.x, xr, fmaf(-lam.y, xi, buv.x));
        float ni = fmaf(lam.x, xi, fmaf( lam.y, xr, buv.y));
        xr = nr; xi = ni;
        out[(((size_t)t * BBAT + b) * HN + h1) * HN + h2] =
            make_float2(xr, xi);
    }
}

// ---------------------------------------------------------------------------
extern "C" void kernel_launch(void* const* d_in, const int* in_sizes, int n_in,
                              void* d_out, int out_size, void* d_ws, size_t ws_size,
                              hipStream_t stream) {
    const float* u     = (const float*)d_in[0];  // (2,1024,1024,1) f32
    const float* delta = (const float*)d_in[1];  // scalar f32
    const float* lre   = (const float*)d_in[2];  // (128,) f32
    const float* lim   = (const float*)d_in[3];  // (128,) f32
    const float* bre   = (const float*)d_in[4];  // (128,1024) f32
    const float* bim   = (const float*)d_in[5];  // (128,1024) f32
    (void)in_sizes; (void)n_in; (void)out_size; (void)ws_size;

    // Workspace layout (floats):
    float*  ws   = (float*)d_ws;
    float*  bbr  = ws;                              // HN*KN
    float*  bbi  = bbr + (size_t)HN * KN;           // HN*KN
    float2* lamb = (float2*)(bbi + (size_t)HN * KN);// HN float2
    float2* bu   = lamb + HN;                       // NCOL*HN float2
    float2* bnd  = bu + (size_t)NCOL * HN;          // NCHUNK*BBAT*HN*HN float2

    // 1) discretization
    {
        int total = HN * KN;
        s5_precompute<<<(total + 255) / 256, 256, 0, stream>>>(
            delta, lre, lim, bre, bim, bbr, bbi, lamb);
    }
    // 2) bu GEMM via WMMA f32 16x16x4 (grid: 128 n-tiles x 8 m-tiles, wave each)
    {
        dim3 grid(NCOL / 16, HN / 16);
        s5_bu_gemm<<<grid, 32, 0, stream>>>(bbr, bbi, u, bu);
    }
    // 3) chunk-boundary states
    s5_scan_boundaries<<<BBAT * HN, HN, 0, stream>>>(lamb, bu, bnd);
    // 4) parallel per-chunk scan with stores
    {
        dim3 grid(BBAT * HN, NCHUNK);
        s5_scan_write<<<grid, HN, 0, stream>>>(lamb, bu, bnd, (float2*)d_out);
    }
}